// ImprovedGraphAutoEncoder_28106265985237
// MI455X (gfx1250) — compile-verified
//
#include <hip/hip_runtime.h>
#include <math.h>

typedef float v2f __attribute__((ext_vector_type(2)));
typedef float v8f __attribute__((ext_vector_type(8)));

#define NEG_SLOPE 0.2f
#define BIG_NEG   (-1e30f)
#define WPB   4          // waves per block
#define WSLOT 4096       // floats per wave-private LDS slot (16KB)
// per-wave LDS offsets (floats)
#define O_HB   0         // 16 x 68 activation buffer (enc h / gat1 out / dec h)
#define O_XL   1088      // 16 x 68 gat1 xl
#define O_XR   2176      // 16 x 68 gat1 xr
#define O_VIS  3264      // 16 x 4
#define O_LAT  3328      // 16 x 4
#define O_XL2  3392      // 16 x 4
#define O_XR2  3456      // 16 x 4
#define O_CB   3520      // 16 x 4 combined
#define O_MB   3584      // 128 mask (2 samples x 64 pairs)
#define O_EA   3712      // 128 edge attr
#define O_EB   3840      // 128 e scores
#define O_AL   3968      // 128 alpha

struct GAEParams {
  const float *x;
  const float *enc_w1, *enc_b1, *enc_w2, *enc_b2, *enc_w3, *enc_b3;
  const float *g1_wl, *g1_bl, *g1_wr, *g1_br, *g1_we, *g1_att, *g1_bias;
  const float *g2_wl, *g2_bl, *g2_wr, *g2_br, *g2_we, *g2_att, *g2_bias;
  const float *skip_w, *skip_b;
  const float *dec_w1, *dec_b1, *dec_w2, *dec_b2, *dec_w3, *dec_b3;
  float *out_xp, *out_rec, *out_vis, *out_adj;
  int rows;   // B * 8
};

__device__ __forceinline__ float wred_sum(float v) {
  #pragma unroll
  for (int off = 1; off < 32; off <<= 1) v += __shfl_xor(v, off, 32);
  return v;
}
__device__ __forceinline__ float g8_sum(float v) {
  v += __shfl_xor(v, 1, 32); v += __shfl_xor(v, 2, 32); v += __shfl_xor(v, 4, 32);
  return v;
}

// 16x64 (LDS, stride 68) @ 64x64 (global, row major) + bias, optional relu,
// result written back into same LDS buffer. Full fp32 via V_WMMA_F32_16X16X4_F32.
__device__ __forceinline__ void gemm64_wmma(float* buf, const float* __restrict__ W,
                                            const float* __restrict__ Bv,
                                            int lane, bool do_relu) {
  const int half = lane >> 4;        // 0: lanes 0-15, 1: lanes 16-31
  const int l15  = lane & 15;
  const int koff = half << 1;        // A-frag K offset per ISA 16x4 layout
  v8f acc[4];
  #pragma unroll
  for (int nt = 0; nt < 4; ++nt) {
    float b = Bv[nt * 16 + l15];
    #pragma unroll
    for (int r = 0; r < 8; ++r) acc[nt][r] = b;
  }
#if __has_builtin(__builtin_amdgcn_wmma_f32_16x16x4_f32)
  for (int kk = 0; kk < 64; kk += 4) {
    const int kc = kk + koff;
    v2f a = *(const v2f*)(buf + l15 * 68 + kc);   // ds_load_b64, conflict-free (stride 68)
    #pragma unroll
    for (int nt = 0; nt < 4; ++nt) {
      const int ncol = nt * 16 + l15;
      v2f bf;
      bf.x = W[kc * 64 + ncol];
      bf.y = W[(kc + 1) * 64 + ncol];
      acc[nt] = __builtin_amdgcn_wmma_f32_16x16x4_f32(
          false, a, false, bf, (short)0, acc[nt], false, false);
    }
  }
  __builtin_amdgcn_wave_barrier();
  #pragma unroll
  for (int nt = 0; nt < 4; ++nt)
    #pragma unroll
    for (int r = 0; r < 8; ++r) {
      float v = acc[nt][r];
      if (do_relu) v = fmaxf(v, 0.0f);
      buf[(r + half * 8) * 68 + nt * 16 + l15] = v;   // C layout: VGPR r -> rows r, r+8
    }
#else
  // VALU fallback (keeps compile green if builtin is absent; histogram will show wmma=0)
  float tmp[32]; int t = 0;
  for (int m = 0; m < 16; ++m)
    for (int cs = 0; cs < 2; ++cs) {
      int c = lane + cs * 32;
      float s = Bv[c];
      for (int k = 0; k < 64; ++k) s += buf[m * 68 + k] * W[k * 64 + c];
      tmp[t++] = do_relu ? fmaxf(s, 0.0f) : s;
    }
  __builtin_amdgcn_wave_barrier();
  t = 0;
  for (int m = 0; m < 16; ++m)
    for (int cs = 0; cs < 2; ++cs) buf[m * 68 + lane + cs * 32] = tmp[t++];
#endif
  __builtin_amdgcn_wave_barrier();
}

__global__ __launch_bounds__(WPB * 32)
void gae_fused_kernel(GAEParams P) {
  __shared__ float smem[WPB * WSLOT];
  const int lane = threadIdx.x & 31;
  const int wv   = threadIdx.x >> 5;
  const int tile = blockIdx.x * WPB + wv;      // 16 rows = 2 samples per wave
  const int nTiles = P.rows >> 4;
  if (tile >= nTiles) return;
  const int row0 = tile << 4;

  float* hb   = smem + wv * WSLOT + O_HB;
  float* xlb  = smem + wv * WSLOT + O_XL;
  float* xrb  = smem + wv * WSLOT + O_XR;
  float* visb = smem + wv * WSLOT + O_VIS;
  float* latb = smem + wv * WSLOT + O_LAT;
  float* xl2b = smem + wv * WSLOT + O_XL2;
  float* xr2b = smem + wv * WSLOT + O_XR2;
  float* cb   = smem + wv * WSLOT + O_CB;
  float* mkb  = smem + wv * WSLOT + O_MB;
  float* eab  = smem + wv * WSLOT + O_EA;
  float* eb   = smem + wv * WSLOT + O_EB;
  float* alb  = smem + wv * WSLOT + O_AL;

  const int cA = lane, cB = lane + 32;
  const int half = lane >> 4, l15 = lane & 15;

  // ---------------- encoder layer1 (3->64, K has only 2 nonzero rows) + xp out ----
  for (int m = 0; m < 16; ++m) {
    float xm = P.x[row0 + m];
    float fi = (float)(m & 7);
    float hA = P.enc_b1[cA] + fi * P.enc_w1[64 + cA] + xm * P.enc_w1[128 + cA];
    float hB = P.enc_b1[cB] + fi * P.enc_w1[64 + cB] + xm * P.enc_w1[128 + cB];
    hb[m * 68 + cA] = fmaxf(hA, 0.0f);
    hb[m * 68 + cB] = fmaxf(hB, 0.0f);
  }
  if (lane < 16) {
    int gr = row0 + lane;
    P.out_xp[gr * 3 + 0] = 0.0f;
    P.out_xp[gr * 3 + 1] = (float)(lane & 7);
    P.out_xp[gr * 3 + 2] = P.x[gr];
  }
  __builtin_amdgcn_wave_barrier();

  // ---------------- encoder layer2 (64->64) on WMMA ----------------
  gemm64_wmma(hb, P.enc_w2, P.enc_b2, lane, true);

  // ---------------- encoder layer3 (64->3) + vis normalization ----------------
  float lat0 = 0.f, lat1 = 0.f, lat2 = 0.f;
  if (lane < 16) {
    float s0 = P.enc_b3[0], s1 = P.enc_b3[1], s2 = P.enc_b3[2];
    for (int n = 0; n < 64; ++n) {
      float h = hb[lane * 68 + n];
      s0 += h * P.enc_w3[n * 3 + 0];
      s1 += h * P.enc_w3[n * 3 + 1];
      s2 += h * P.enc_w3[n * 3 + 2];
    }
    lat0 = s0; lat1 = s1; lat2 = s2;
  }
  float vis0, vis1, vis2;
  {
    float m0 = g8_sum(lat0) * 0.125f; float v0 = lat0 - m0;
    float m1 = g8_sum(lat1) * 0.125f; float v1 = lat1 - m1;
    float m2 = g8_sum(lat2) * 0.125f; float v2 = lat2 - m2;
    float sd0 = sqrtf(g8_sum(v0 * v0) * (1.0f / 7.0f)) + 1e-8f;
    float sd1 = sqrtf(g8_sum(v1 * v1) * (1.0f / 7.0f)) + 1e-8f;
    float sd2 = sqrtf(g8_sum(v2 * v2) * (1.0f / 7.0f)) + 1e-8f;
    vis0 = v0 / sd0; vis1 = v1 / sd1; vis2 = v2 / sd2;
  }
  if (lane < 16) {
    latb[lane * 4 + 0] = lat0; latb[lane * 4 + 1] = lat1; latb[lane * 4 + 2] = lat2;
    visb[lane * 4 + 0] = vis0; visb[lane * 4 + 1] = vis1; visb[lane * 4 + 2] = vis2;
    int gr = row0 + lane;
    P.out_vis[gr * 3 + 0] = vis0; P.out_vis[gr * 3 + 1] = vis1; P.out_vis[gr * 3 + 2] = vis2;
  }
  __builtin_amdgcn_wave_barrier();

  // ---------------- Gabriel graph: 128 (sample,i,j) pairs, 4 per lane ----------------
  float adjv[4], distv[4];
  float nE0 = 0.f, nE1 = 0.f, sD0 = 0.f, sD1 = 0.f;
  #pragma unroll
  for (int q = 0; q < 4; ++q) {
    int p = lane + q * 32;
    int s = p >> 6, i = (p >> 3) & 7, j = p & 7;
    const float* vi = visb + (s * 8 + i) * 4;
    const float* vj = visb + (s * 8 + j) * 4;
    float vix = vi[0], viy = vi[1], viz = vi[2];
    float vjx = vj[0], vjy = vj[1], vjz = vj[2];
    float mx = 0.5f * (vix + vjx), my = 0.5f * (viy + vjy), mz = 0.5f * (viz + vjz);
    float r2 = (vix - mx) * (vix - mx) + (viy - my) * (viy - my) + (viz - mz) * (viz - mz);
    bool viol = false;
    #pragma unroll
    for (int k = 0; k < 8; ++k) {
      if (k == i || k == j) continue;
      const float* vk = visb + (s * 8 + k) * 4;
      float dx = vk[0] - mx, dy = vk[1] - my, dz = vk[2] - mz;
      viol = viol || (dx * dx + dy * dy + dz * dz < r2);
    }
    bool a = (i != j) && !viol;
    float dx = vix - vjx, dy = viy - vjy, dz = viz - vjz;
    float dist = sqrtf(dx * dx + dy * dy + dz * dz + 1e-30f);
    adjv[q] = a ? 1.0f : 0.0f; distv[q] = dist;
    if (q < 2) { nE0 += adjv[q]; sD0 += a ? dist : 0.0f; }
    else       { nE1 += adjv[q]; sD1 += a ? dist : 0.0f; }
    P.out_adj[(tile * 2 + s) * 64 + i * 8 + j] = adjv[q];
  }
  nE0 = wred_sum(nE0); sD0 = wred_sum(sD0);
  nE1 = wred_sum(nE1); sD1 = wred_sum(sD1);
  float mA0 = sD0 / fmaxf(nE0, 1.0f);
  float mA1 = sD1 / fmaxf(nE1, 1.0f);
  #pragma unroll
  for (int q = 0; q < 4; ++q) {
    int p = lane + q * 32;
    int s = p >> 6, i = (p >> 3) & 7, j = p & 7;
    mkb[p] = (i == j || adjv[q] > 0.5f) ? 1.0f : 0.0f;
    eab[p] = (i == j) ? (s ? mA1 : mA0) : distv[q];
  }
  __builtin_amdgcn_wave_barrier();

  // ---------------- GAT1: xl/xr transforms (3->64) ----------------
  for (int m = 0; m < 16; ++m) {
    float l0 = latb[m * 4 + 0], l1 = latb[m * 4 + 1], l2 = latb[m * 4 + 2];
    xlb[m * 68 + cA] = P.g1_bl[cA] + l0 * P.g1_wl[cA] + l1 * P.g1_wl[64 + cA] + l2 * P.g1_wl[128 + cA];
    xlb[m * 68 + cB] = P.g1_bl[cB] + l0 * P.g1_wl[cB] + l1 * P.g1_wl[64 + cB] + l2 * P.g1_wl[128 + cB];
    xrb[m * 68 + cA] = P.g1_br[cA] + l0 * P.g1_wr[cA] + l1 * P.g1_wr[64 + cA] + l2 * P.g1_wr[128 + cA];
    xrb[m * 68 + cB] = P.g1_br[cB] + l0 * P.g1_wr[cB] + l1 * P.g1_wr[64 + cB] + l2 * P.g1_wr[128 + cB];
  }
  __builtin_amdgcn_wave_barrier();

  // ---------------- GAT1: edge scores e[i,j] = sum_c lrelu(xr+xl+ea*we)*att ----------
  #pragma unroll
  for (int q = 0; q < 4; ++q) {
    int p = lane + q * 32;
    int s = p >> 6, i = (p >> 3) & 7, j = p & 7;
    int ri = s * 8 + i, rj = s * 8 + j;
    float ea = eab[p];
    float acc = 0.0f;
    for (int c = 0; c < 64; ++c) {
      float t = xrb[ri * 68 + c] + xlb[rj * 68 + c] + ea * P.g1_we[c];
      t = (t >= 0.0f) ? t : NEG_SLOPE * t;
      acc += t * P.g1_att[c];
    }
    eb[p] = (mkb[p] > 0.5f) ? acc : BIG_NEG;
  }
  __builtin_amdgcn_wave_barrier();
  // row-wise masked softmax (16 rows, one per lane 0-15)
  if (lane < 16) {
    int base = lane * 8;
    float mx = eb[base];
    for (int j = 1; j < 8; ++j) mx = fmaxf(mx, eb[base + j]);
    float ex[8]; float sum = 0.0f;
    for (int j = 0; j < 8; ++j) { ex[j] = expf(eb[base + j] - mx); sum += ex[j]; }
    float inv = 1.0f / sum;
    for (int j = 0; j < 8; ++j) alb[base + j] = ex[j] * inv * mkb[base + j];
  }
  __builtin_amdgcn_wave_barrier();
  // aggregation + bias + relu -> hb (GAT1 output)
  for (int m = 0; m < 16; ++m) {
    int s = m >> 3; int ab = m * 8;
    float oA = P.g1_bias[cA], oB = P.g1_bias[cB];
    #pragma unroll
    for (int j = 0; j < 8; ++j) {
      float a = alb[ab + j];
      oA += a * xlb[(s * 8 + j) * 68 + cA];
      oB += a * xlb[(s * 8 + j) * 68 + cB];
    }
    hb[m * 68 + cA] = fmaxf(oA, 0.0f);
    hb[m * 68 + cB] = fmaxf(oB, 0.0f);
  }
  __builtin_amdgcn_wave_barrier();

  // ---------------- GAT2: xl2/xr2 (64->3), lanes 0-15 do xl2, 16-31 do xr2 ----------
  {
    int m = l15;
    const float* W  = half ? P.g2_wr : P.g2_wl;
    const float* Bb = half ? P.g2_br : P.g2_bl;
    float* ob = half ? xr2b : xl2b;
    float s0 = Bb[0], s1 = Bb[1], s2 = Bb[2];
    for (int n = 0; n < 64; ++n) {
      float h = hb[m * 68 + n];
      s0 += h * W[n * 3 + 0]; s1 += h * W[n * 3 + 1]; s2 += h * W[n * 3 + 2];
    }
    ob[m * 4 + 0] = s0; ob[m * 4 + 1] = s1; ob[m * 4 + 2] = s2;
  }
  __builtin_amdgcn_wave_barrier();
  // GAT2 edge scores
  #pragma unroll
  for (int q = 0; q < 4; ++q) {
    int p = lane + q * 32;
    int s = p >> 6, i = (p >> 3) & 7, j = p & 7;
    int ri = s * 8 + i, rj = s * 8 + j;
    float ea = eab[p];
    float acc = 0.0f;
    #pragma unroll
    for (int l = 0; l < 3; ++l) {
      float t = xr2b[ri * 4 + l] + xl2b[rj * 4 + l] + ea * P.g2_we[l];
      t = (t >= 0.0f) ? t : NEG_SLOPE * t;
      acc += t * P.g2_att[l];
    }
    eb[p] = (mkb[p] > 0.5f) ? acc : BIG_NEG;
  }
  __builtin_amdgcn_wave_barrier();
  if (lane < 16) {
    int base = lane * 8;
    float mx = eb[base];
    for (int j = 1; j < 8; ++j) mx = fmaxf(mx, eb[base + j]);
    float ex[8]; float sum = 0.0f;
    for (int j = 0; j < 8; ++j) { ex[j] = expf(eb[base + j] - mx); sum += ex[j]; }
    float inv = 1.0f / sum;
    for (int j = 0; j < 8; ++j) alb[base + j] = ex[j] * inv * mkb[base + j];
  }
  __builtin_amdgcn_wave_barrier();
  // GAT2 aggregation + skip -> combined
  if (lane < 16) {
    int m = lane; int s = m >> 3; int ab = m * 8;
    float o0 = P.g2_bias[0], o1 = P.g2_bias[1], o2 = P.g2_bias[2];
    #pragma unroll
    for (int j = 0; j < 8; ++j) {
      float a = alb[ab + j];
      o0 += a * xl2b[(s * 8 + j) * 4 + 0];
      o1 += a * xl2b[(s * 8 + j) * 4 + 1];
      o2 += a * xl2b[(s * 8 + j) * 4 + 2];
    }
    float l0 = latb[m * 4 + 0], l1 = latb[m * 4 + 1], l2 = latb[m * 4 + 2];
    cb[m * 4 + 0] = o0 + P.skip_b[0] + l0 * P.skip_w[0] + l1 * P.skip_w[3] + l2 * P.skip_w[6];
    cb[m * 4 + 1] = o1 + P.skip_b[1] + l0 * P.skip_w[1] + l1 * P.skip_w[4] + l2 * P.skip_w[7];
    cb[m * 4 + 2] = o2 + P.skip_b[2] + l0 * P.skip_w[2] + l1 * P.skip_w[5] + l2 * P.skip_w[8];
  }
  __builtin_amdgcn_wave_barrier();

  // ---------------- decoder: 3->64 (VALU), 64->64 (WMMA), 64->3 (VALU) --------------
  for (int m = 0; m < 16; ++m) {
    float c0 = cb[m * 4 + 0], c1 = cb[m * 4 + 1], c2 = cb[m * 4 + 2];
    float hA = P.dec_b1[cA] + c0 * P.dec_w1[cA] + c1 * P.dec_w1[64 + cA] + c2 * P.dec_w1[128 + cA];
    float hB = P.dec_b1[cB] + c0 * P.dec_w1[cB] + c1 * P.dec_w1[64 + cB] + c2 * P.dec_w1[128 + cB];
    hb[m * 68 + cA] = fmaxf(hA, 0.0f);
    hb[m * 68 + cB] = fmaxf(hB, 0.0f);
  }
  __builtin_amdgcn_wave_barrier();
  gemm64_wmma(hb, P.dec_w2, P.dec_b2, lane, true);
  if (lane < 16) {
    float s0 = P.dec_b3[0], s1 = P.dec_b3[1], s2 = P.dec_b3[2];
    for (int n = 0; n < 64; ++n) {
      float h = hb[lane * 68 + n];
      s0 += h * P.dec_w3[n * 3 + 0];
      s1 += h * P.dec_w3[n * 3 + 1];
      s2 += h * P.dec_w3[n * 3 + 2];
    }
    int gr = row0 + lane;
    P.out_rec[gr * 3 + 0] = s0; P.out_rec[gr * 3 + 1] = s1; P.out_rec[gr * 3 + 2] = s2;
  }
}

extern "C" void kernel_launch(void* const* d_in, const int* in_sizes, int n_in,
                              void* d_out, int out_size, void* d_ws, size_t ws_size,
                              hipStream_t stream) {
  (void)n_in; (void)out_size; (void)d_ws; (void)ws_size;
  GAEParams P;
  P.x      = (const float*)d_in[0];
  P.enc_w1 = (const float*)d_in[1];  P.enc_b1 = (const float*)d_in[2];
  P.enc_w2 = (const float*)d_in[3];  P.enc_b2 = (const float*)d_in[4];
  P.enc_w3 = (const float*)d_in[5];  P.enc_b3 = (const float*)d_in[6];
  P.g1_wl  = (const float*)d_in[7];  P.g1_bl  = (const float*)d_in[8];
  P.g1_wr  = (const float*)d_in[9];  P.g1_br  = (const float*)d_in[10];
  P.g1_we  = (const float*)d_in[11]; P.g1_att = (const float*)d_in[12];
  P.g1_bias= (const float*)d_in[13];
  P.g2_wl  = (const float*)d_in[14]; P.g2_bl  = (const float*)d_in[15];
  P.g2_wr  = (const float*)d_in[16]; P.g2_br  = (const float*)d_in[17];
  P.g2_we  = (const float*)d_in[18]; P.g2_att = (const float*)d_in[19];
  P.g2_bias= (const float*)d_in[20];
  P.skip_w = (const float*)d_in[21]; P.skip_b = (const float*)d_in[22];
  P.dec_w1 = (const float*)d_in[23]; P.dec_b1 = (const float*)d_in[24];
  P.dec_w2 = (const float*)d_in[25]; P.dec_b2 = (const float*)d_in[26];
  P.dec_w3 = (const float*)d_in[27]; P.dec_b3 = (const float*)d_in[28];

  const int rows = in_sizes[0];           // B * 8 node-rows
  float* out = (float*)d_out;
  P.out_xp  = out;                        // [rows,3]
  P.out_rec = out + (size_t)rows * 3;     // [rows,3]
  P.out_vis = out + (size_t)rows * 6;     // [rows,3]
  P.out_adj = out + (size_t)rows * 9;     // [B,8,8] = rows*8 floats (0/1)
  P.rows = rows;

  const int tiles  = rows / 16;           // 2 samples per wave
  const int blocks = (tiles + WPB - 1) / WPB;
  gae_fused_kernel<<<blocks, WPB * 32, 0, stream>>>(P);
}